// EfficientGlobalPointer_21088289423438
// MI455X (gfx1250) — compile-verified
//
#include <hip/hip_runtime.h>

typedef __attribute__((ext_vector_type(2))) float v2f;
typedef __attribute__((ext_vector_type(8))) float v8f;

#define NEG_BIG 1.0e12f
#define ROWS_TOTAL 8192   // 16 * 512
#define H_IN 1024
#define H_X 128

__device__ __forceinline__ v8f wmma_f32(v2f a, v2f b, v8f c) {
  // V_WMMA_F32_16X16X4_F32 : D(16x16 f32) = A(16x4 f32) x B(4x16 f32) + C
  return __builtin_amdgcn_wmma_f32_16x16x4_f32(false, a, false, b, (short)0, c,
                                               false, false);
}

// -------------------------------------------------------------------------
// Stage 1: x = inputs@W1 + b1 (WMMA), RoPE -> q,k ; bias = (x@W2+b2)/2
// One block per 64-row slab (rows = b*512+m flattened; 512 % 64 == 0 so a
// block never straddles a batch). 4 waves, each owns 16 rows x 128 cols.
// -------------------------------------------------------------------------
__global__ __launch_bounds__(128)
void egp_stage1(const float* __restrict__ inp,
                const float* __restrict__ W1,
                const float* __restrict__ b1,
                const float* __restrict__ W2,
                const float* __restrict__ b2,
                float* __restrict__ qo,    // [8192][64]
                float* __restrict__ ko,    // [8192][64]
                float* __restrict__ bmo,   // [12][8192]  odd channels (m-axis)
                float* __restrict__ bno) { // [12][8192]  even channels (n-axis)
  __shared__ float xbuf[64 * 132];  // stride 132: bank-spread + 16B aligned rows

  const int t    = threadIdx.x;
  const int wave = t >> 5;
  const int lane = t & 31;
  const int half = lane >> 4;
  const int N    = lane & 15;
  const int kb   = half * 2;
  const int rg0  = blockIdx.x * 64;

  v8f acc[8] = {};

  // ---- GEMM1: K=1024 in chunks of 128 staged through LDS ----
  for (int kc0 = 0; kc0 < H_IN; kc0 += 128) {
#pragma unroll
    for (int i = 0; i < 16; ++i) {           // 64x128 floats cooperative load
      int f4 = i * 128 + t;
      int r  = f4 >> 5;
      int c4 = f4 & 31;
      float4 v = *(const float4*)&inp[(size_t)(rg0 + r) * H_IN + kc0 + c4 * 4];
      *(float4*)&xbuf[r * 132 + c4 * 4] = v;
    }
    __syncthreads();
    const int arow = wave * 16 + N;          // A-operand row for this lane
#pragma unroll 4
    for (int ks = 0; ks < 128; ks += 4) {
      v2f a = *(const v2f*)&xbuf[arow * 132 + ks + kb];
      const float* w0 = &W1[(size_t)(kc0 + ks + kb) * H_X];
#pragma unroll
      for (int j = 0; j < 8; ++j) {          // 8 n-subtiles of 16
        v2f bb;
        bb.x = w0[j * 16 + N];
        bb.y = w0[H_X + j * 16 + N];
        acc[j] = wmma_f32(a, bb, acc[j]);
      }
    }
    __syncthreads();
  }

  // ---- spill x (+b1) row-major into xbuf (reuse staging LDS) ----
#pragma unroll
  for (int j = 0; j < 8; ++j) {
    int col = j * 16 + N;
    float bias1 = b1[col];
#pragma unroll
    for (int i = 0; i < 8; ++i) {
      int ml = wave * 16 + half * 8 + i;     // C/D layout: M = i + half*8
      xbuf[ml * 132 + col] = acc[j][i] + bias1;
    }
  }
  __syncthreads();

  // ---- RoPE: qw = x[::2], kw = x[1::2]; pair (2j,2j+1), angle pos*inv[j] ----
  for (int task = t; task < 64 * 64; task += 128) {
    int r  = task >> 6;
    int d  = task & 63;
    int jj = d >> 1;
    float pos   = (float)((rg0 + r) & 511);
    float theta = pos * __expf(-0.2878231366f * (float)jj); // 10000^(-2j/64)
    float s, c;
    __sincosf(theta, &s, &c);
    const float* xr = &xbuf[r * 132];
    float q0 = xr[4 * jj],     k0 = xr[4 * jj + 1];
    float q1 = xr[4 * jj + 2], k1 = xr[4 * jj + 3];
    float qv = (d & 1) ? (q1 * c + q0 * s) : (q0 * c - q1 * s);
    float kv = (d & 1) ? (k1 * c + k0 * s) : (k0 * c - k1 * s);
    qo[(size_t)(rg0 + r) * 64 + d] = qv;
    ko[(size_t)(rg0 + r) * 64 + d] = kv;
  }

  // ---- bias = (x@W2 + b2)/2 ; even channel -> n-axis, odd -> m-axis ----
  for (int task = t; task < 64 * 24; task += 128) {
    int r = task / 24;
    int o = task - r * 24;
    float s = b2[o];
    const float* xr = &xbuf[r * 132];
#pragma unroll 4
    for (int kx = 0; kx < H_X; ++kx) s = fmaf(xr[kx], W2[kx * 24 + o], s);
    s *= 0.5f;
    int h = o >> 1;
    float* dst = (o & 1) ? bmo : bno;
    dst[h * ROWS_TOTAL + rg0 + r] = s;
  }
}

// -------------------------------------------------------------------------
// Stage 2: out[b,h,m,n] = (q_m . k_n)/8 + bn[h,n] + bm[h,m] - masks
// Block = 64x64 (m x n) tile of one batch; 4 waves, wave w owns rows
// m0+16w..+15, accumulating 4 chained-K WMMA tiles across n. qk computed
// once, broadcast to 12 heads via float4 coalesced stores (201 MB total).
// -------------------------------------------------------------------------
__global__ __launch_bounds__(128)
void egp_logits(const float* __restrict__ q,
                const float* __restrict__ k,
                const float* __restrict__ bm,
                const float* __restrict__ bn,
                const float* __restrict__ amask,
                float* __restrict__ out) {
  __shared__ float qs[64 * 68];   // stride 68: conflict-free column reads
  __shared__ float ks[64 * 68];
  __shared__ float bms[12 * 64];
  __shared__ float bns[12 * 64];
  __shared__ float ams[64];
  __shared__ float ans[64];

  const int t    = threadIdx.x;
  const int wave = t >> 5;
  const int lane = t & 31;
  const int half = lane >> 4;
  const int N    = lane & 15;
  const int kb   = half * 2;

  const int n0 = blockIdx.x * 64;
  const int m0 = blockIdx.y * 64;
  const int b  = blockIdx.z;
  const int rbase = b * 512;

  // ---- stage q/k tiles + biases + mask slices into LDS ----
#pragma unroll
  for (int i = 0; i < 8; ++i) {
    int f4 = i * 128 + t;
    int r  = f4 >> 4;
    int c4 = f4 & 15;
    *(float4*)&qs[r * 68 + c4 * 4] =
        *(const float4*)&q[(size_t)(rbase + m0 + r) * 64 + c4 * 4];
    *(float4*)&ks[r * 68 + c4 * 4] =
        *(const float4*)&k[(size_t)(rbase + n0 + r) * 64 + c4 * 4];
  }
  for (int i = t; i < 12 * 64; i += 128) {
    int h = i >> 6, r = i & 63;
    bms[i] = bm[h * ROWS_TOTAL + rbase + m0 + r];
    bns[i] = bn[h * ROWS_TOTAL + rbase + n0 + r];
  }
  if (t < 64) ams[t] = amask[rbase + m0 + t];
  else        ans[t - 64] = amask[rbase + n0 + (t - 64)];
  __syncthreads();

  // ---- qk: K=64 as 16 chained 16x16x4 WMMAs, 4 n-subtiles per wave ----
  v8f acc[4] = {};
  const int arow = wave * 16 + N;
#pragma unroll
  for (int kk = 0; kk < 64; kk += 4) {
    v2f a = *(const v2f*)&qs[arow * 68 + kk + kb];
#pragma unroll
    for (int j = 0; j < 4; ++j) {
      v2f bb = *(const v2f*)&ks[(j * 16 + N) * 68 + kk + kb];
      acc[j] = wmma_f32(a, bb, acc[j]);
    }
  }
  __syncthreads();               // qs/ks reads done; reuse qs as store staging

  // ---- fold scale + attention mask + causal mask once (head-independent) ----
  float* ov = qs;
#pragma unroll
  for (int j = 0; j < 4; ++j) {
    int nl = j * 16 + N;
    float an = ans[nl];
#pragma unroll
    for (int i = 0; i < 8; ++i) {
      int ml = wave * 16 + half * 8 + i;
      float val = acc[j][i] * 0.125f;            // 1/sqrt(64)
      val -= (1.0f - ams[ml] * an) * NEG_BIG;    // attention mask
      if ((n0 + nl) < (m0 + ml)) val -= NEG_BIG; // tril(k=-1) causal mask
      ov[ml * 68 + nl] = val;
    }
  }
  __syncthreads();

  // ---- 12-head broadcast with float4 coalesced stores ----
#pragma unroll
  for (int h = 0; h < 12; ++h) {
    size_t obase = ((size_t)(b * 12 + h)) * 512 * 512;
#pragma unroll
    for (int rr = 0; rr < 8; ++rr) {
      int ml = wave * 16 + rr * 2 + half;
      float4 v   = *(const float4*)&ov[ml * 68 + N * 4];
      float  bmh = bms[h * 64 + ml];
      float4 bn4 = *(const float4*)&bns[h * 64 + N * 4];
      float4 o4;
      o4.x = v.x + bmh + bn4.x;
      o4.y = v.y + bmh + bn4.y;
      o4.z = v.z + bmh + bn4.z;
      o4.w = v.w + bmh + bn4.w;
      *(float4*)&out[obase + (size_t)(m0 + ml) * 512 + n0 + N * 4] = o4;
    }
  }
}

extern "C" void kernel_launch(void* const* d_in, const int* in_sizes, int n_in,
                              void* d_out, int out_size, void* d_ws, size_t ws_size,
                              hipStream_t stream) {
  const float* inp   = (const float*)d_in[0];
  const float* amask = (const float*)d_in[1];
  const float* W1    = (const float*)d_in[2];
  const float* b1    = (const float*)d_in[3];
  const float* W2    = (const float*)d_in[4];
  const float* b2    = (const float*)d_in[5];
  float* out = (float*)d_out;

  float* w   = (float*)d_ws;                    // ~4.98 MB total
  float* qo  = w;
  float* ko  = qo  + (size_t)ROWS_TOTAL * 64;
  float* bmo = ko  + (size_t)ROWS_TOTAL * 64;
  float* bno = bmo + (size_t)12 * ROWS_TOTAL;

  egp_stage1<<<128, 128, 0, stream>>>(inp, W1, b1, W2, b2, qo, ko, bmo, bno);
  egp_logits<<<dim3(8, 8, 16), 128, 0, stream>>>(qo, ko, bmo, bno, amask, out);
}